// InpaintContextAttentionUnit5edge_52965536694874
// MI455X (gfx1250) — compile-verified
//
#include <hip/hip_runtime.h>

typedef float f2 __attribute__((ext_vector_type(2)));
typedef float f4 __attribute__((ext_vector_type(4)));
typedef float f8 __attribute__((ext_vector_type(8)));

#define B_   8
#define H_   256
#define W_   512
#define C_   16
#define NP   8      // pooled H
#define PW   256    // pooled W
#define PELEMS (B_ * NP * PW * C_)   // 262144 floats per intermediate

// ---------------------------------------------------------------------------
// Kernel 1: 32x2 average pool  fm[8,256,512,16] -> P[8,8,256,16]
// One thread per (b, y, x, 4-channel group); float4 loads, fully vectorized.
// ---------------------------------------------------------------------------
__global__ void iac_pool_kernel(const float* __restrict__ fm, float* __restrict__ P) {
  int tid = blockIdx.x * blockDim.x + threadIdx.x;       // 65536 threads
  int c4 = tid & 3;
  int x  = (tid >> 2) & (PW - 1);
  int y  = (tid >> 10) & (NP - 1);
  int b  = tid >> 13;

  f4 acc = {};
  const float* base = fm + (((size_t)b * H_ + (size_t)y * 32) * W_ + (size_t)2 * x) * C_ + 4 * c4;
  #pragma unroll 4
  for (int i = 0; i < 32; ++i) {
    const float* r = base + (size_t)i * (W_ * C_);
    acc += *(const f4*)(r);
    acc += *(const f4*)(r + C_);
  }
  acc *= (1.0f / 64.0f);
  *(f4*)(P + (((size_t)(b * NP + y) * PW + x) * C_) + 4 * c4) = acc;
}

// ---------------------------------------------------------------------------
// Kernel 2: masked 5x5 conv (rows 0 and 4 only) + bias + relu via
// V_WMMA_F32_16X16X4_F32.  One wave (32 lanes) computes a 16(x-positions) x
// 16(filters) output tile; K-loop = 10 taps x 4 channel-chunks = 40 WMMAs.
// Blocks 0..1023  -> row_op (offsets: dy=kr-2, dx=kc-2)
// Blocks 1024..2047 -> col_op (offsets swapped: dy=kc-2, dx=kr-2)
// A-fragment layout (32-bit 16x4): lane l -> M=l%16, K pair {0,1} for lanes
// 0-15 / {2,3} for lanes 16-31  => contiguous float2 load from channel dim.
// ---------------------------------------------------------------------------
__global__ __launch_bounds__(32) void iac_conv_wmma_kernel(
    const float* __restrict__ P, const float* __restrict__ ker,
    const float* __restrict__ bias,
    float* __restrict__ rop, float* __restrict__ cop) {
  int g = blockIdx.x;                 // 0..2047
  int is_col = (g >= 1024) ? 1 : 0;
  int t  = g & 1023;
  int xt = t & 15;
  int y  = (t >> 4) & 7;
  int b  = t >> 7;

  int lane = threadIdx.x;
  int m  = lane & 15;                 // M index (and N index for B/D frags)
  int kh = lane >> 4;                 // K-half select
  int x0 = xt * 16;

  f8 acc = {};
  #pragma unroll
  for (int kr = 0; kr <= 4; kr += 4) {
    #pragma unroll
    for (int kc = 0; kc < 5; ++kc) {
      int oy = is_col ? (kc - 2) : (kr - 2);
      int ox = is_col ? (kr - 2) : (kc - 2);
      int yy = y + oy;
      int xx = x0 + m + ox;
      bool valid = (yy >= 0) && (yy < NP) && (xx >= 0) && (xx < PW);
      long aoff = (((long)(b * NP + yy) * PW + xx) * C_);
      const float* wp = ker + (size_t)((kr * 5 + kc) * C_) * 16 + m; // HWIO, +chan*16
      #pragma unroll
      for (int k0 = 0; k0 < 16; k0 += 4) {
        int chan = k0 + 2 * kh;       // this lane's K indices: chan, chan+1
        f2 zero = {};
        f2 a = valid ? *(const f2*)(P + aoff + chan) : zero;
        f2 bb;
        bb.x = wp[chan * 16];         // B[K=chan][N=m]
        bb.y = wp[chan * 16 + 16];    // B[K=chan+1][N=m]
        acc = __builtin_amdgcn_wmma_f32_16x16x4_f32(
            /*neg_a=*/false, a, /*neg_b=*/false, bb,
            /*c_mod=*/(short)0, acc, /*reuse_a=*/false, /*reuse_b=*/false);
      }
    }
  }

  float bv = bias[m];
  float* outp = is_col ? cop : rop;
  #pragma unroll
  for (int v = 0; v < 8; ++v) {
    int row = v + 8 * kh;             // D: M = v + 8*(lane/16), N = lane%16
    float val = acc[v] + bv;
    val = val > 0.0f ? val : 0.0f;
    outp[((size_t)(b * NP + y) * PW + (x0 + row)) * C_ + m] = val;
  }
}

// ---------------------------------------------------------------------------
// Kernel 3: fused bilinear upsample (8x256 -> 256x512, jax half-pixel
// convention == edge-clamped sampling) + subtract + concat.
// out[b,h,w, 0:16]  = fm
// out[b,h,w,16:32]  = fm - bilinear(row_op)
// out[b,h,w,32:48]  = fm - bilinear(col_op)
// One thread per output pixel; row_op/col_op (1 MB each) stay L2-resident.
// ---------------------------------------------------------------------------
__global__ void iac_fuse_kernel(const float* __restrict__ fm,
                                const float* __restrict__ rop,
                                const float* __restrict__ cop,
                                float* __restrict__ out) {
  int idx = blockIdx.x * blockDim.x + threadIdx.x;      // 1,048,576 threads
  int w = idx & (W_ - 1);
  int h = (idx >> 9) & (H_ - 1);
  int b = idx >> 17;

  float yin = (h + 0.5f) * (1.0f / 32.0f) - 0.5f;       // 8 -> 256
  float yf  = floorf(yin);
  int   iy0 = (int)yf;
  float fy  = yin - yf;
  int y0c = min(max(iy0, 0), NP - 1);
  int y1c = min(max(iy0 + 1, 0), NP - 1);

  float xin = (w + 0.5f) * 0.5f - 0.5f;                 // 256 -> 512
  float xf  = floorf(xin);
  int   ix0 = (int)xf;
  float fx  = xin - xf;
  int x0c = min(max(ix0, 0), PW - 1);
  int x1c = min(max(ix0 + 1, 0), PW - 1);

  float w00 = (1.0f - fy) * (1.0f - fx);
  float w01 = (1.0f - fy) * fx;
  float w10 = fy * (1.0f - fx);
  float w11 = fy * fx;

  size_t o00 = ((size_t)(b * NP + y0c) * PW + x0c) * C_;
  size_t o01 = ((size_t)(b * NP + y0c) * PW + x1c) * C_;
  size_t o10 = ((size_t)(b * NP + y1c) * PW + x0c) * C_;
  size_t o11 = ((size_t)(b * NP + y1c) * PW + x1c) * C_;
  size_t fmb = ((size_t)(b * H_ + h) * W_ + w) * C_;
  size_t ob  = ((size_t)(b * H_ + h) * W_ + w) * (3 * C_);

  #pragma unroll
  for (int c = 0; c < C_; c += 4) {
    f4 f = *(const f4*)(fm + fmb + c);
    f4 r = w00 * *(const f4*)(rop + o00 + c) + w01 * *(const f4*)(rop + o01 + c)
         + w10 * *(const f4*)(rop + o10 + c) + w11 * *(const f4*)(rop + o11 + c);
    f4 q = w00 * *(const f4*)(cop + o00 + c) + w01 * *(const f4*)(cop + o01 + c)
         + w10 * *(const f4*)(cop + o10 + c) + w11 * *(const f4*)(cop + o11 + c);
    *(f4*)(out + ob + c)            = f;
    *(f4*)(out + ob + C_ + c)       = f - r;
    *(f4*)(out + ob + 2 * C_ + c)   = f - q;
  }
}

// ---------------------------------------------------------------------------
extern "C" void kernel_launch(void* const* d_in, const int* in_sizes, int n_in,
                              void* d_out, int out_size, void* d_ws, size_t ws_size,
                              hipStream_t stream) {
  const float* fm   = (const float*)d_in[0];   // [8,256,512,16]
  const float* ker  = (const float*)d_in[1];   // [5,5,16,16] HWIO
  const float* bias = (const float*)d_in[2];   // [16]
  float* out = (float*)d_out;                  // [8,256,512,48]

  float* P   = (float*)d_ws;                   // pooled   [8,8,256,16]
  float* rop = P + PELEMS;                     // row_op   [8,8,256,16]
  float* cop = rop + PELEMS;                   // col_op   [8,8,256,16]

  iac_pool_kernel<<<256, 256, 0, stream>>>(fm, P);
  iac_conv_wmma_kernel<<<2048, 32, 0, stream>>>(P, ker, bias, rop, cop);
  iac_fuse_kernel<<<4096, 256, 0, stream>>>(fm, rop, cop, out);
}